// CoExDispProcessor_82403242541677
// MI455X (gfx1250) — compile-verified
//
#include <hip/hip_runtime.h>
#include <stdint.h>

#ifndef USE_ASYNC_LDS
#define USE_ASYNC_LDS 1
#endif

namespace {
constexpr int D   = 48;
constexpr int Hq  = 128;          // quarter-res h
constexpr int Wq  = 240;          // quarter-res w
constexpr int HWq = Hq * Wq;      // 30720
constexpr int Ho  = 4 * Hq;       // 512
constexpr int Wo  = 4 * Wq;       // 960
constexpr int HWo = Ho * Wo;      // 491520
constexpr int TX  = 64;           // output tile width  (full-res)
constexpr int TY  = 16;           // output tile height (full-res)
}

// ---------------------------------------------------------------------------
// Stage 1: per-pixel top-2 over D=48 + 2-way softmax soft-argmax.
// One thread per 4 consecutive pixels: every plane load is global_load_b128
// (512 B contiguous per wave per plane), store is b128.
// ---------------------------------------------------------------------------
__global__ __launch_bounds__(256) void disp4_topk_kernel(
    const float* __restrict__ cost, float* __restrict__ disp4, int total4) {
  int idx4 = blockIdx.x * 256 + threadIdx.x;   // flat over b*h*w/4
  if (idx4 >= total4) return;
  int b    = idx4 / (HWq / 4);
  int pix4 = idx4 - b * (HWq / 4);
  const float* p = cost + (size_t)b * D * HWq + (size_t)pix4 * 4;

  float v1[4], v2[4];
  int   i1[4], i2[4];
#pragma unroll
  for (int j = 0; j < 4; ++j) {
    v1[j] = -INFINITY; v2[j] = -INFINITY; i1[j] = 0; i2[j] = 0;
  }

#pragma unroll 3
  for (int dd = 0; dd < D; dd += 4) {
    // speculative prefetch 8 planes ahead (silently dropped if unmapped)
    __builtin_prefetch(p + (size_t)(dd + 8) * HWq, 0, 0);
#pragma unroll
    for (int jd = 0; jd < 4; ++jd) {
      const int d = dd + jd;
      const float4 v4 = *reinterpret_cast<const float4*>(p + (size_t)d * HWq);
      const float ve[4] = {v4.x, v4.y, v4.z, v4.w};
#pragma unroll
      for (int j = 0; j < 4; ++j) {
        float v = ve[j];
        bool g1 = v > v1[j];
        bool g2 = v > v2[j];
        float nv2 = g1 ? v1[j] : (g2 ? v : v2[j]);
        int   ni2 = g1 ? i1[j] : (g2 ? d : i2[j]);
        v1[j] = g1 ? v : v1[j];
        i1[j] = g1 ? d : i1[j];
        v2[j] = nv2;
        i2[j] = ni2;
      }
    }
  }

  float4 o;
  float od[4];
#pragma unroll
  for (int j = 0; j < 4; ++j) {
    float e2 = __expf(v2[j] - v1[j]);               // vals sorted desc
    od[j] = fmaf(e2, (float)i2[j], (float)i1[j]) / (1.0f + e2);
  }
  o.x = od[0]; o.y = od[1]; o.z = od[2]; o.w = od[3];
  reinterpret_cast<float4*>(disp4)[idx4] = o;
}

// ---------------------------------------------------------------------------
// Stage 2: 9-ch spx softmax * 3x3-unfolded 4x-nearest-upsampled disp4, *4.
// 64x16 output tile / 256-thread block; 18x6 disp4 halo in LDS (zero-padded)
// staged with gfx1250 async global->LDS copy. One thread per 4 consecutive
// x's: the 4 outputs share one disp4 cell, so the 9 patch values are shared,
// spg channel loads are b128, output store is b128.
// ---------------------------------------------------------------------------
__global__ __launch_bounds__(256) void upfeat_kernel(
    const float* __restrict__ spg, const float* __restrict__ disp4,
    float* __restrict__ out) {
  __shared__ float sm[6 * 18];    // halo: (TY/4 + 2) x (TX/4 + 2)

  const int tid = threadIdx.x;
  const int bx  = blockIdx.x, by = blockIdx.y, b = blockIdx.z;
  const int wi0 = bx * (TX / 4);  // disp4 tile origin (16 wide)
  const int hi0 = by * (TY / 4);  // (4 tall)

  // zero the padding first, then async-copy the in-range halo elements
  if (tid < 108) sm[tid] = 0.0f;
  __syncthreads();
  if (tid < 108) {
    int lh = tid / 18, lw = tid - lh * 18;
    int hi = hi0 - 1 + lh, wi = wi0 - 1 + lw;
    if ((unsigned)hi < (unsigned)Hq && (unsigned)wi < (unsigned)Wq) {
      const float* gp = disp4 + (size_t)b * HWq + hi * Wq + wi;
#if USE_ASYNC_LDS
      uint32_t lds = (uint32_t)(uintptr_t)&sm[tid];
      asm volatile("global_load_async_to_lds_b32 %0, %1, off"
                   :: "v"(lds), "v"(gp) : "memory");
#else
      sm[tid] = *gp;
#endif
    }
  }
#if USE_ASYNC_LDS
  asm volatile("s_wait_asynccnt 0" ::: "memory");
#endif
  __syncthreads();

  const int txg = tid & 15;            // x-group 0..15 (4 outputs each)
  const int ty  = tid >> 4;            // row 0..15
  const int x0  = bx * TX + txg * 4;   // 16B-aligned
  const int y   = by * TY + ty;
  const int lh  = ty >> 2;             // disp4 row within tile: 0..3

  // 9 shared patch values (LDS broadcast: all 4 outputs use the same cell)
  float pk[9];
#pragma unroll
  for (int ki = 0; ki < 3; ++ki)
#pragma unroll
    for (int kj = 0; kj < 3; ++kj)
      pk[ki * 3 + kj] = sm[(lh + ki) * 18 + txg + kj];

  const float* sp = spg + (size_t)b * 9 * HWo + (size_t)y * Wo + x0;
  float4 sv[9];
#pragma unroll
  for (int k = 0; k < 9; ++k)
    sv[k] = *reinterpret_cast<const float4*>(sp + (size_t)k * HWo);

  float4 m = sv[0];
#pragma unroll
  for (int k = 1; k < 9; ++k) {
    m.x = fmaxf(m.x, sv[k].x);
    m.y = fmaxf(m.y, sv[k].y);
    m.z = fmaxf(m.z, sv[k].z);
    m.w = fmaxf(m.w, sv[k].w);
  }

  float4 sum = {0.f, 0.f, 0.f, 0.f};
  float4 acc = {0.f, 0.f, 0.f, 0.f};
#pragma unroll
  for (int k = 0; k < 9; ++k) {
    float ex = __expf(sv[k].x - m.x);
    float ey = __expf(sv[k].y - m.y);
    float ez = __expf(sv[k].z - m.z);
    float ew = __expf(sv[k].w - m.w);
    sum.x += ex; sum.y += ey; sum.z += ez; sum.w += ew;
    acc.x = fmaf(ex, pk[k], acc.x);
    acc.y = fmaf(ey, pk[k], acc.y);
    acc.z = fmaf(ez, pk[k], acc.z);
    acc.w = fmaf(ew, pk[k], acc.w);
  }

  float4 o;
  o.x = 4.0f * acc.x / sum.x;
  o.y = 4.0f * acc.y / sum.y;
  o.z = 4.0f * acc.z / sum.z;
  o.w = 4.0f * acc.w / sum.w;
  *reinterpret_cast<float4*>(out + (size_t)b * HWo + (size_t)y * Wo + x0) = o;
}

extern "C" void kernel_launch(void* const* d_in, const int* in_sizes, int n_in,
                              void* d_out, int out_size, void* d_ws, size_t ws_size,
                              hipStream_t stream) {
  const float* cost = (const float*)d_in[0];   // [b,1,48,128,240] f32
  const float* spg  = (const float*)d_in[1];   // [b,9,512,960]    f32
  float* out   = (float*)d_out;                // [b,512,960]      f32
  float* disp4 = (float*)d_ws;                 // [b,128,240] scratch (983 KB)

  const int b      = in_sizes[0] / (D * HWq);
  const int total4 = b * HWq / 4;

  disp4_topk_kernel<<<(total4 + 255) / 256, 256, 0, stream>>>(cost, disp4, total4);

  dim3 g2(Wo / TX, Ho / TY, b);                // (15, 32, b)
  upfeat_kernel<<<g2, 256, 0, stream>>>(spg, disp4, out);
}